// ExponentialDecay_23708219474744
// MI455X (gfx1250) — compile-verified
//
#include <hip/hip_runtime.h>

typedef float v2f __attribute__((ext_vector_type(2)));
typedef float v8f __attribute__((ext_vector_type(8)));

#define EMA_ALPHA 0.9f
#define T_TOTAL   1024
#define TB        16
#define NBLK      (T_TOTAL / TB)      // 64 blocks of 16 timesteps
#define C_TOTAL   (32 * 1024)         // 32768 independent channels

// One wave owns 16 channels. Each 16-step block is D = A(16x16) x X(16x16)
// via four chained V_WMMA_F32_16X16X4_F32, plus a rank-1 carry term.
__global__ __launch_bounds__(256) void ema_scan_wmma_kernel(
    const float* __restrict__ x,        // [1024][32768]
    const float* __restrict__ state,    // [32768]
    float* __restrict__ out,            // [1024][32768]
    float* __restrict__ final_state)    // [32768]
{
    const int lane = threadIdx.x & 31;
    const int wave = threadIdx.x >> 5;
    const int n    = lane & 15;         // column / row index within 16
    const int half = lane >> 4;         // 0: lanes 0-15, 1: lanes 16-31

    const int c0  = (blockIdx.x * 8 + wave) * 16;   // 16 channels per wave
    const int col = c0 + n;

    // alpha powers 0..16 (constant-folds after unroll)
    float pw[TB + 1];
    pw[0] = 1.0f;
#pragma unroll
    for (int i = 1; i <= TB; ++i) pw[i] = pw[i - 1] * EMA_ALPHA;

    // A matrix, 4 K-chunks of 16x4 f32.
    // ISA layout: row M = lane%16; within chunk ch, VGPR j holds K = 4*ch + 2*half + j.
    v2f A[4];
#pragma unroll
    for (int ch = 0; ch < 4; ++ch) {
#pragma unroll
        for (int j = 0; j < 2; ++j) {
            const int k = 4 * ch + 2 * half + j;
            A[ch][j] = (n >= k) ? (1.0f - EMA_ALPHA) * pw[n - k] : 0.0f;
        }
    }

    // Carry decay per C/D VGPR: row M = r + 8*half -> alpha^(M+1)
    float decay[8];
#pragma unroll
    for (int r = 0; r < 8; ++r) decay[r] = pw[r + 8 * half] * EMA_ALPHA;

    float s = state[col];                       // running carry for this column

    const float* xp = x + col;
    float*       op = out + col;

    for (int b = 0; b < NBLK; ++b) {
        const int t0 = b * TB;

        // Load X tile: B[k][n] = x[t0+k][col], same K-per-VGPR striping as A.
        v2f B[4];
#pragma unroll
        for (int ch = 0; ch < 4; ++ch) {
#pragma unroll
            for (int j = 0; j < 2; ++j) {
                const int k = 4 * ch + 2 * half + j;
                B[ch][j] = xp[(size_t)(t0 + k) * C_TOTAL];
            }
        }

        // Prefetch next block's rows for this lane (global_prefetch_b8, no counter).
        if (b + 1 < NBLK) {
#pragma unroll
            for (int ch = 0; ch < 4; ++ch) {
#pragma unroll
                for (int j = 0; j < 2; ++j) {
                    const int k = 4 * ch + 2 * half + j;
                    __builtin_prefetch(&xp[(size_t)(t0 + TB + k) * C_TOTAL], 0, 3);
                }
            }
        }

        // D = A x X  (four chained f32 WMMAs over K = 16)
        v8f acc = {};
#pragma unroll
        for (int ch = 0; ch < 4; ++ch) {
            acc = __builtin_amdgcn_wmma_f32_16x16x4_f32(
                /*neg_a=*/false, A[ch], /*neg_b=*/false, B[ch],
                /*c_mod=*/(short)0, acc, /*reuse_a=*/false, /*reuse_b=*/false);
        }

        // Add carried state, store 16 rows (coalesced b32 per row-pair).
        float o7 = 0.0f;
#pragma unroll
        for (int r = 0; r < 8; ++r) {
            const float v = acc[r] + decay[r] * s;
            op[(size_t)(t0 + r + 8 * half) * C_TOTAL] = v;
            if (r == 7) o7 = v;
        }

        // Next carry = row 15 (VGPR7, upper half), broadcast by column.
        s = __shfl(o7, 16 + n, 32);
    }

    final_state[col] = s;
}

extern "C" void kernel_launch(void* const* d_in, const int* in_sizes, int n_in,
                              void* d_out, int out_size, void* d_ws, size_t ws_size,
                              hipStream_t stream) {
    (void)in_sizes; (void)n_in; (void)d_ws; (void)ws_size; (void)out_size;

    const float* x     = (const float*)d_in[0];   // (1024, 32, 1024) f32
    const float* state = (const float*)d_in[1];   // (32, 1024) f32

    float* out = (float*)d_out;                               // first output: all states
    float* fin = out + (size_t)T_TOTAL * C_TOTAL;             // second output: final state

    // 2048 waves total: 256 blocks x 8 waves, 16 channels per wave.
    const int blocks = C_TOTAL / (16 * 8);                    // 256
    ema_scan_wmma_kernel<<<blocks, 256, 0, stream>>>(x, state, out, fin);
}